// QuantumLayer_23828478558269
// MI455X (gfx1250) — compile-verified
//
#include <hip/hip_runtime.h>
#include <cstdint>
#include <cstring>

// ---------------------------------------------------------------------------
// Quantum circuit simulator for MI455X (gfx1250, wave32).
//   state: d_ws, 512 samples x 65536 complex64 (float2) = 256 MB.
//   Passes: 1 init (RX product state) + 12 LDS-tiled Rot-gate passes
//           (CNOT layers folded into GF(2) index maps, zero data movement)
//           + 1 WMMA-f32 reduction pass for <Z_q>.
// ---------------------------------------------------------------------------

#define NQ      16
#define NLAYERS 6
#define BATCH   512
#define TBITS   13          // LDS tile = 2^13 amplitudes = 64 KB
#define TILEN   (1 << TBITS)

typedef float v2f __attribute__((ext_vector_type(2)));
typedef float v8f __attribute__((ext_vector_type(8)));

struct PassDesc {
    int      layer;
    int      ngates;        // <= 13
    int      nf;            // 13 - ngates : unit-vector fill basis slots
    uint16_t basis[13];     // physical-index XOR basis for u-bits 0..12
    uint16_t coset[3];      // physical-index XOR vectors for 3 coset bits
    uint8_t  qubit[13];     // logical qubit (weights index)
    uint16_t mphys[13];     // physical parity mask (row of F)
};

struct RedDesc { uint32_t m[16]; };   // measurement parity masks per qubit

// ---------------------------------------------------------------------------
// Pass 0: build post-RX product state.
//   amp(p) = (-i)^popcount(p) * prod_j (bit_j(p) ? sin(x[15-j]/2) : cos(..))
// ---------------------------------------------------------------------------
__global__ __launch_bounds__(256)
void init_state_kernel(float2* __restrict__ state, const float* __restrict__ x)
{
    __shared__ float cA[16], sA[16];
    const int tid   = threadIdx.x;
    const int b     = blockIdx.x >> 2;
    const int chunk = blockIdx.x & 3;
    if (tid < 16) {
        // bit position j == tid  <->  qubit q = 15 - j  (reference big-endian)
        float xv = x[b * 16 + (15 - tid)];
        cA[tid] = __cosf(0.5f * xv) * 0.0f + cosf(0.5f * xv); // precise cosf
        sA[tid] = sinf(0.5f * xv);
    }
    __syncthreads();
    const size_t sbase = ((size_t)b) << 16;
    for (int e = 0; e < 64; ++e) {
        unsigned p = (unsigned)chunk * 16384u + (unsigned)e * 256u + tid;
        float mag = 1.0f;
#pragma unroll
        for (int j = 0; j < 16; ++j) mag *= ((p >> j) & 1u) ? sA[j] : cA[j];
        int k = __popc(p) & 3;                      // (-i)^popcount phase
        float re = (k == 0) ?  mag : ((k == 2) ? -mag : 0.0f);
        float im = (k == 1) ? -mag : ((k == 3) ?  mag : 0.0f);
        state[sbase + p] = make_float2(re, im);
    }
}

// ---------------------------------------------------------------------------
// Gate pass: stage a 2^13-amplitude coset in LDS, apply up to 13 Rot gates
// (each gate's pair partner is a single u-space bit by basis construction).
// ---------------------------------------------------------------------------
__global__ __launch_bounds__(256)
void gate_pass_kernel(float2* __restrict__ state,
                      const float* __restrict__ weights, PassDesc pd)
{
    __shared__ float2 tile[TILEN];                  // 64 KB of 320 KB WGP LDS
    const int tid = threadIdx.x;
    const int b   = blockIdx.x >> 3;
    const int g   = blockIdx.x & 7;

    unsigned cbase = 0;
    if (g & 1) cbase ^= pd.coset[0];
    if (g & 2) cbase ^= pd.coset[1];
    if (g & 4) cbase ^= pd.coset[2];

    // physical index for u = tid (low 8 u-bits)
    unsigned pt = cbase;
#pragma unroll
    for (int i = 0; i < 8; ++i) if ((tid >> i) & 1) pt ^= pd.basis[i];
    const size_t sbase = ((size_t)b) << 16;

    // ---- load tile (u = e*256 + tid) ----
#pragma unroll
    for (int e = 0; e < 32; ++e) {
        unsigned pe = pt;
#pragma unroll
        for (int i = 0; i < 5; ++i) if ((e >> i) & 1) pe ^= pd.basis[8 + i];
        tile[e * 256 + tid] = state[sbase + pe];
    }
    __syncthreads();

    // ---- apply gates ----
    for (int j = 0; j < pd.ngates; ++j) {
        const int q = pd.qubit[j];
        const float* w = weights + ((size_t)pd.layer * 16 + q) * 3;
        const float phi = w[0], th = w[1], om = w[2];
        float st, ct;  sincosf(0.5f * th, &st, &ct);
        float sap, cap; sincosf(0.5f * (phi + om), &sap, &cap);
        float sam, cam; sincosf(0.5f * (phi - om), &sam, &cam);
        // Rot = RZ(om) RY(th) RZ(phi)
        const float u00r =  cap * ct, u00i = -sap * ct;
        const float u01r = -cam * st, u01i = -sam * st;
        const float u10r =  cam * st, u10i = -sam * st;
        const float u11r =  cap * ct, u11i =  sap * ct;

        const unsigned m = pd.mphys[j];
        unsigned pm = 0;
#pragma unroll
        for (int i = 0; i < 13; ++i)
            pm |= (unsigned)(__popc(pd.basis[i] & m) & 1) << i;
        const unsigned cpar = (unsigned)(__popc(cbase & m) & 1);

        const int      pos     = pd.nf + j;         // partner bit in u-space
        const unsigned lowmask = (1u << pos) - 1u;

#pragma unroll
        for (int k = 0; k < 16; ++k) {              // 4096 pairs / 256 threads
            unsigned v  = (unsigned)k * 256u + tid; // 12-bit pair index
            unsigned u  = ((v & ~lowmask) << 1) | (v & lowmask);
            unsigned u2 = u | (1u << pos);
            unsigned b0 = ((unsigned)__popc(u & pm) & 1u) ^ cpar;
            float2 A = tile[u], B = tile[u2];
            float a0r, a0i, a1r, a1i;
            if (b0) { a0r = B.x; a0i = B.y; a1r = A.x; a1i = A.y; }
            else    { a0r = A.x; a0i = A.y; a1r = B.x; a1i = B.y; }
            float y0r = u00r*a0r - u00i*a0i + u01r*a1r - u01i*a1i;
            float y0i = u00r*a0i + u00i*a0r + u01r*a1i + u01i*a1r;
            float y1r = u10r*a0r - u10i*a0i + u11r*a1r - u11i*a1i;
            float y1i = u10r*a0i + u10i*a0r + u11r*a1i + u11i*a1r;
            float2 o0 = make_float2(y0r, y0i), o1 = make_float2(y1r, y1i);
            if (b0) { tile[u] = o1; tile[u2] = o0; }
            else    { tile[u] = o0; tile[u2] = o1; }
        }
        __syncthreads();
    }

    // ---- store tile ----
#pragma unroll
    for (int e = 0; e < 32; ++e) {
        unsigned pe = pt;
#pragma unroll
        for (int i = 0; i < 5; ++i) if ((e >> i) & 1) pe ^= pd.basis[8 + i];
        state[sbase + pe] = tile[e * 256 + tid];
    }
}

// ---------------------------------------------------------------------------
// Reduction pass: <Z_q> = sum_p |amp(p)|^2 * (-1)^parity(p & m_q)
// Low-4-bit sign factor handled by V_WMMA_F32_16X16X4_F32 (exact f32 matmul),
// high-12-bit sign factor applied per D element, then deterministic tree sum.
// One block (8 waves) per sample.
// ---------------------------------------------------------------------------
__global__ __launch_bounds__(256)
void reduce_kernel(const float2* __restrict__ state, float* __restrict__ out,
                   RedDesc rd)
{
    __shared__ float red[8 * 16];
    const int tid  = threadIdx.x;
    const int lane = tid & 31;
    const int wave = tid >> 5;
    const int b    = blockIdx.x;
    const size_t sbase = ((size_t)b) << 16;

    const unsigned n     = lane & 15;              // N index (qubit)
    const unsigned mq    = rd.m[n];
    const int      khalf = (lane < 16) ? 0 : 2;    // K half per A/B layout

    // B fragments (signs over the low 4 bits of p), one per K-chunk
    v2f bf[4];
#pragma unroll
    for (int kc = 0; kc < 4; ++kc) {
        unsigned pl0 = (unsigned)kc * 4u + khalf;
        bf[kc].x = (__popc(pl0 & mq & 0xFu) & 1) ? -1.0f : 1.0f;
        bf[kc].y = (__popc((pl0 + 1u) & mq & 0xFu) & 1) ? -1.0f : 1.0f;
    }
    const unsigned mh = mq >> 4;

    float acc = 0.0f;
    const float4* st4 = (const float4*)state;
    for (int t = wave; t < 256; t += 8) {          // 256 ph-tiles per sample
        const int phbase = t * 16;
        v8f c = {0, 0, 0, 0, 0, 0, 0, 0};
#pragma unroll
        for (int kc = 0; kc < 4; ++kc) {
            // A: M = lane%16 (ph row), K = kc*4 + khalf (+1 in .y)
            unsigned p0 = ((unsigned)(phbase + (int)n) << 4)
                        + (unsigned)kc * 4u + khalf;
            float4 s = st4[(sbase + p0) >> 1];     // two adjacent complex amps
            v2f a;
            a.x = s.x * s.x + s.y * s.y;           // prob(p0)
            a.y = s.z * s.z + s.w * s.w;           // prob(p0+1)
            c = __builtin_amdgcn_wmma_f32_16x16x4_f32(
                    false, a, false, bf[kc], (short)0, c, false, false);
        }
        // D layout: VGPR r, lane L -> M = r + (L<16?0:8), N = L%16
#pragma unroll
        for (int r = 0; r < 8; ++r) {
            unsigned M  = (unsigned)r + ((lane < 16) ? 0u : 8u);
            unsigned ph = (unsigned)phbase + M;
            float v = c[r];
            acc += (__popc(ph & mh) & 1) ? -v : v; // high-bit sign factor
        }
    }
    acc += __shfl_xor(acc, 16, 32);                // fold lanes 16-31 into 0-15
    if (lane < 16) red[wave * 16 + lane] = acc;
    __syncthreads();
    if (tid < 16) {
        float s = 0.0f;
#pragma unroll
        for (int w = 0; w < 8; ++w) s += red[w * 16 + tid];
        out[b * 16 + tid] = s;                     // exps[b][q]
    }
}

// ---------------------------------------------------------------------------
// Host-side GF(2) machinery (pure, deterministic, constant-only).
// F rows as 16-bit masks: y_i = parity(Frow[i] & p).
// ---------------------------------------------------------------------------
static void gf2_inv(const uint32_t a_in[16], uint32_t inv[16])
{
    uint32_t a[16];
    for (int i = 0; i < 16; ++i) { a[i] = a_in[i]; inv[i] = 1u << i; }
    for (int col = 0; col < 16; ++col) {
        int piv = -1;
        for (int r = col; r < 16; ++r)
            if ((a[r] >> col) & 1u) { piv = r; break; }
        uint32_t ta = a[col];  a[col]  = a[piv];  a[piv]  = ta;
        uint32_t ti = inv[col]; inv[col] = inv[piv]; inv[piv] = ti;
        for (int r = 0; r < 16; ++r)
            if (r != col && ((a[r] >> col) & 1u)) { a[r] ^= a[col]; inv[r] ^= inv[col]; }
    }
}

static void build_pass(int layer, const int* qubits, int k,
                       const uint32_t Frow[16], const uint32_t Finv[16],
                       PassDesc& pd)
{
    pd.layer = layer; pd.ngates = k; pd.nf = 13 - k;
    uint16_t masks[13];
    for (int j = 0; j < k; ++j) {
        int q = qubits[j], beta = 15 - q;           // logical bit position
        uint16_t s = 0;
        for (int i = 0; i < 16; ++i)                // column beta of F^-1
            s = (uint16_t)(s | (((Finv[i] >> beta) & 1u) << i));
        masks[j]    = s;
        pd.qubit[j] = (uint8_t)q;
        pd.mphys[j] = (uint16_t)(Frow[beta] & 0xFFFFu);
    }
    // pivot selection (prefer high bits -> low bits stay free for coalescing)
    uint16_t red[13]; int pivot[13]; bool used[16] = {false};
    for (int j = 0; j < k; ++j) {
        uint16_t v = masks[j];
        for (;;) {
            int h = 31 - __builtin_clz((uint32_t)v); // v != 0 (independent cols)
            int who = -1;
            for (int i = 0; i < j; ++i) if (pivot[i] == h) { who = i; break; }
            if (who < 0) { pivot[j] = h; red[j] = v; used[h] = true; break; }
            v = (uint16_t)(v ^ red[who]);
        }
    }
    int nf = 0, nc = 0;
    for (int pos = 0; pos < 16; ++pos) {
        if (used[pos]) continue;
        if (nf < pd.nf) pd.basis[nf++] = (uint16_t)(1u << pos);
        else            pd.coset[nc++] = (uint16_t)(1u << pos);
    }
    for (int j = 0; j < k; ++j) pd.basis[pd.nf + j] = masks[j];
}

// ---------------------------------------------------------------------------
extern "C" void kernel_launch(void* const* d_in, const int* in_sizes, int n_in,
                              void* d_out, int out_size, void* d_ws,
                              size_t ws_size, hipStream_t stream)
{
    (void)in_sizes; (void)n_in; (void)out_size; (void)ws_size;
    const float* d_x = (const float*)d_in[0];      // (512,16) f32
    const float* d_w = (const float*)d_in[1];      // (6,16,3) f32
    float2*      st  = (float2*)d_ws;              // 512 * 2^16 complex64
    float*       out = (float*)d_out;              // (512,16) f32

    // Pass 0: |0..0> + RX embedding as a product state (one full write).
    init_state_kernel<<<BATCH * 4, 256, 0, stream>>>(st, d_x);

    uint32_t Frow[16];
    for (int i = 0; i < 16; ++i) Frow[i] = 1u << i;

    for (int l = 0; l < NLAYERS; ++l) {
        uint32_t Finv[16];
        gf2_inv(Frow, Finv);
        PassDesc p1, p2;
        int q1[13]; for (int j = 0; j < 13; ++j) q1[j] = j;
        int q2[3] = {13, 14, 15};
        build_pass(l, q1, 13, Frow, Finv, p1);
        build_pass(l, q2,  3, Frow, Finv, p2);
        gate_pass_kernel<<<BATCH * 8, 256, 0, stream>>>(st, d_w, p1);
        gate_pass_kernel<<<BATCH * 8, 256, 0, stream>>>(st, d_w, p2);
        // Fold this layer's CNOT ring into F (zero data movement):
        // CNOT(q, (q+r)%16): logical bit(15-t) ^= bit(15-c), sequentially.
        int r = (l % (NQ - 1)) + 1;
        for (int q = 0; q < NQ; ++q) {
            int t = (q + r) % NQ;
            Frow[15 - t] ^= Frow[15 - q];
        }
    }

    RedDesc rd;
    for (int q = 0; q < 16; ++q) rd.m[q] = Frow[15 - q];
    reduce_kernel<<<BATCH, 256, 0, stream>>>(st, out, rd);
}